// PointCloud2LaserScanLoss_86947317940406
// MI455X (gfx1250) — compile-verified
//
#include <hip/hip_runtime.h>

typedef __attribute__((ext_vector_type(2))) float v2f;
typedef __attribute__((ext_vector_type(8))) float v8f;

#define B_            8
#define N_            4096
#define M_            4096
#define ROWS_PER_BLK  128                   // 8 waves * 16 pred rows
#define TILES_PER_B   (N_ / ROWS_PER_BLK)   // 32
#define NBLOCKS       (B_ * TILES_PER_B)    // 256
#define CHUNK         2048                  // targets staged per pass (32 KB LDS)
#define BIGVAL        1e30f

// One block: batch b, 128 predicted rows. Targets staged in LDS pre-swizzled
// for the WMMA B operand: float4 {-2tx, -2ty, |t|^2|mask, 0} per target.
// Lower 16 lanes (B rows K=0,1) read bytes [0,8); upper lanes (K=2,3) read
// bytes [8,16) -> one uniform ds_load_b64 per tile, zero divergence.
// V_WMMA_F32_16X16X4_F32 with A=[px,py,1,0] then yields |t|^2 - 2 p.t
// (masked) directly in the accumulator; min-reduce gives 1-NN distance.
__global__ __launch_bounds__(256) void pc_nn_kernel(
    const float* __restrict__ pc,   // [B,N,2] predicted_coords
    const int*   __restrict__ pp,   // [B]     predicted_points
    const float* __restrict__ tc,   // [B,M,2] target_coords
    const int*   __restrict__ tp,   // [B]     target_points
    float*       __restrict__ ws)   // [NBLOCKS] per-block partial sums
{
    __shared__ float4 sB[CHUNK];            // 32 KB, pre-swizzled B operand
    __shared__ float  sRow[ROWS_PER_BLK];
    __shared__ float  sRed[ROWS_PER_BLK];

    const int blk     = blockIdx.x;
    const int b       = blk / TILES_PER_B;
    const int tile    = blk % TILES_PER_B;
    const int rowBase = tile * ROWS_PER_BLK;
    const int t       = threadIdx.x;
    const int nt      = tp[b];

    const int  wave  = t >> 5;
    const int  lane  = t & 31;
    const int  nidx  = lane & 15;
    const int  half  = (lane >> 4) & 1;     // 0: K=0,1 rows; 1: K=2,3 rows

    // ---- A matrix (16x4 f32): lanes 0-15 -> K=0,1 ; lanes 16-31 -> K=2,3 ----
    const float2* pcv = (const float2*)(pc + (size_t)b * N_ * 2);
    v2f a;
    if (half == 0) {
        float2 p = pcv[rowBase + wave * 16 + nidx];
        a[0] = p.x; a[1] = p.y;             // K=0,1 : px, py
    } else {
        a[0] = 1.0f; a[1] = 0.0f;           // K=2,3 : 1, 0
    }

    v8f acc;
#pragma unroll
    for (int v = 0; v < 8; ++v) acc[v] = BIGVAL;

    const float2* tcv = (const float2*)(tc + (size_t)b * M_ * 2);
    const float2* sB2 = (const float2*)sB;

    for (int c0 = 0; c0 < M_; c0 += CHUNK) {
        __syncthreads();                    // previous chunk fully consumed
        // ---- stage + swizzle CHUNK targets (coalesced, b128 LDS stores) ----
        for (int j = t; j < CHUNK; j += 256) {
            const int g = c0 + j;
            float2 cc = tcv[g];
            float tn = (g < nt) ? (cc.x * cc.x + cc.y * cc.y) : BIGVAL;
            float4 e; e.x = -2.0f * cc.x; e.y = -2.0f * cc.y; e.z = tn; e.w = 0.0f;
            sB[j] = e;
        }
        __syncthreads();

        // ---- 128 tiles: 1 uniform ds_load_b64 + 1 WMMA + fused mins each ----
#pragma unroll 4
        for (int jt = 0; jt < CHUNK / 16; ++jt) {
            float2 bv = sB2[(((jt << 4) + nidx) << 1) | half];
            v2f bb; bb[0] = bv.x; bb[1] = bv.y;
            v8f c = {};
            c = __builtin_amdgcn_wmma_f32_16x16x4_f32(
                    false, a, false, bb, (short)0, c, false, false);
#pragma unroll
            for (int v = 0; v < 8; ++v) acc[v] = fminf(acc[v], c[v]);
        }
    }

    // ---- min over the 16 lanes of each half (xor masks < 16 stay in-half) ----
#pragma unroll
    for (int m = 1; m < 16; m <<= 1) {
#pragma unroll
        for (int v = 0; v < 8; ++v)
            acc[v] = fminf(acc[v], __shfl_xor(acc[v], m, 32));
    }
    // C/D layout: row M = v (lower half) / v+8 (upper half)
    if (lane == 0) {
#pragma unroll
        for (int v = 0; v < 8; ++v) sRow[wave * 16 + v] = acc[v];
    } else if (lane == 16) {
#pragma unroll
        for (int v = 0; v < 8; ++v) sRow[wave * 16 + 8 + v] = acc[v];
    }
    __syncthreads();

    // ---- finalize rows: add |p|^2 back, mask invalid preds, block-sum ----
    const int np = pp[b];
    if (t < ROWS_PER_BLK) {
        float val = 0.0f;
        const int i = rowBase + t;
        if (i < np) {
            float2 p = pcv[i];
            float sq = sRow[t] + p.x * p.x + p.y * p.y;
            val = fmaxf(sq, 0.0f);
        }
        sRed[t] = val;
    }
    __syncthreads();
    if (t == 0) {
        float s = 0.0f;
        for (int k = 0; k < ROWS_PER_BLK; ++k) s += sRed[k];  // fixed order
        ws[blk] = s;
    }
}

__global__ void pc_finalize(const int* __restrict__ pp,
                            const int* __restrict__ tp,
                            const float* __restrict__ ws,
                            float* __restrict__ out)
{
    if (threadIdx.x == 0 && blockIdx.x == 0) {
        float coord = 0.0f, pts = 0.0f;
        for (int b = 0; b < B_; ++b) {
            float s = 0.0f;
            for (int k = 0; k < TILES_PER_B; ++k) s += ws[b * TILES_PER_B + k];
            coord += s / ((float)pp[b] * 2.0f);
            float dd = ((float)pp[b] - (float)tp[b]) / (float)N_;
            pts += dd * dd;
        }
        coord *= (1.0f / B_);
        pts   *= (1.0f / B_);
        out[0] = coord + 0.1f * pts;   // total_loss
        out[1] = coord;                // coord_loss
        out[2] = pts;                  // points_loss
    }
}

extern "C" void kernel_launch(void* const* d_in, const int* in_sizes, int n_in,
                              void* d_out, int out_size, void* d_ws, size_t ws_size,
                              hipStream_t stream) {
    const float* pc = (const float*)d_in[0];   // predicted_coords [B,N,2]
    const int*   pp = (const int*)  d_in[1];   // predicted_points [B]
    const float* tc = (const float*)d_in[2];   // target_coords    [B,M,2]
    const int*   tp = (const int*)  d_in[3];   // target_points    [B]
    float* ws  = (float*)d_ws;                 // NBLOCKS floats
    float* out = (float*)d_out;                // 3 floats

    pc_nn_kernel<<<NBLOCKS, 256, 0, stream>>>(pc, pp, tc, tp, ws);
    pc_finalize<<<1, 32, 0, stream>>>(pp, tp, ws, out);
}